// GCNBlock_52218212385050
// MI455X (gfx1250) — compile-verified
//
#include <hip/hip_runtime.h>

#define N_NODES 50000
#define N_EDGES 800000
#define DIM 128
#define ATILE_STRIDE 132  // pad 16x128 LDS tile to avoid bank conflicts (4-dword skew per row)

typedef __attribute__((ext_vector_type(2))) float v2f;
typedef __attribute__((ext_vector_type(8))) float v8f;

// ---------------------------------------------------------------------------
// 1) Zero the output (used as the scatter accumulator) and the degree buffers.
// ---------------------------------------------------------------------------
__global__ void __launch_bounds__(256) gcn_zero_kernel(float* out, float* deg) {
    int i = blockIdx.x * blockDim.x + threadIdx.x;
    int stride = gridDim.x * blockDim.x;
    for (int j = i; j < N_NODES * DIM; j += stride) out[j] = 0.0f;
    for (int j = i; j < 2 * N_NODES; j += stride) deg[j] = 0.0f;
}

// ---------------------------------------------------------------------------
// 2) Degrees: deg_out[src]++ , deg_in[dst]++  (f32 atomics: exact for counts)
// ---------------------------------------------------------------------------
__global__ void __launch_bounds__(256) gcn_degree_kernel(const int* __restrict__ ei,
                                                         float* deg_out, float* deg_in) {
    int e = blockIdx.x * blockDim.x + threadIdx.x;
    if (e >= N_EDGES) return;
    atomicAdd(&deg_out[ei[e]], 1.0f);
    atomicAdd(&deg_in[ei[N_EDGES + e]], 1.0f);
}

// ---------------------------------------------------------------------------
// 3) norm = rsqrt(max(deg,1)) in place; also transpose W -> Wt[n][k] so the
//    WMMA B fragments are contiguous float2 loads along K.
// ---------------------------------------------------------------------------
__global__ void __launch_bounds__(256) gcn_norm_transpose_kernel(float* deg_out, float* deg_in,
                                                                 const float* __restrict__ W,
                                                                 float* __restrict__ Wt) {
    int i = blockIdx.x * blockDim.x + threadIdx.x;
    if (i < N_NODES) {
        deg_out[i] = rsqrtf(fmaxf(deg_out[i], 1.0f));
        deg_in[i]  = rsqrtf(fmaxf(deg_in[i], 1.0f));
    }
    if (i < DIM * DIM) {
        int k = i >> 7, n = i & (DIM - 1);
        Wt[n * DIM + k] = W[i];  // W is row-major [k][n]
    }
}

// ---------------------------------------------------------------------------
// 4) Scatter: one wave per edge; lane handles 4 features (32*4 = 128).
//    agg[dst] += x[src] * norm_src[src]   via global_atomic_add_f32.
//    Working set (x + agg ~ 51 MB) is L2-resident on MI455X (192 MB L2).
// ---------------------------------------------------------------------------
__global__ void __launch_bounds__(256) gcn_scatter_kernel(const float* __restrict__ x,
                                                          const int* __restrict__ ei,
                                                          const float* __restrict__ norm_src,
                                                          float* agg) {
    int lane = threadIdx.x & 31;
    int wave = threadIdx.x >> 5;
    int e = blockIdx.x * 8 + wave;
    if (e >= N_EDGES) return;
    int src = ei[e];
    int dst = ei[N_EDGES + e];
    float ns = norm_src[src];
    float4 xv = *(const float4*)(x + src * DIM + lane * 4);
    float* ap = agg + dst * DIM + lane * 4;
    atomicAdd(ap + 0, xv.x * ns);
    atomicAdd(ap + 1, xv.y * ns);
    atomicAdd(ap + 2, xv.z * ns);
    atomicAdd(ap + 3, xv.w * ns);
}

// ---------------------------------------------------------------------------
// 5) GEMM: out = relu((agg * norm_dst) @ W + b), in place on `io`.
//    Block = one 16-row stripe (50000 = 3125*16). 8 waves = 8 N-tiles of 16.
//    A tile staged through LDS (norm_dst folded in); in-place is safe because
//    all reads of io rows happen before __syncthreads(), all writes after.
//    Per wave: 32 x v_wmma_f32_16x16x4_f32 over K=128.
// ---------------------------------------------------------------------------
__global__ void __launch_bounds__(256) gcn_wmma_gemm_kernel(const float* __restrict__ Wt,
                                                            const float* __restrict__ bias,
                                                            const float* __restrict__ norm_dst,
                                                            float* io) {
    __shared__ float atile[16 * ATILE_STRIDE];

    int tid  = threadIdx.x;
    int lane = tid & 31;
    int wave = tid >> 5;          // 0..7 -> N tile
    int m0   = blockIdx.x * 16;

    // Stage A-tile: atile[r][c] = io[m0+r][c] * norm_dst[m0+r]  (coalesced)
    for (int i = tid; i < 16 * DIM; i += 256) {
        int r = i >> 7;
        int c = i & (DIM - 1);
        atile[r * ATILE_STRIDE + c] = io[(m0 + r) * DIM + c] * norm_dst[m0 + r];
    }
    __syncthreads();

    // Fragment indexing per CDNA5 ISA (f32 16x16x4):
    //   A: lane holds M = lane&15, K = 2*(lane>>4) + r  (r = vgpr index 0..1)
    //   B: lane holds N = lane&15, K = 2*(lane>>4) + r
    //   C/D: lane holds N = lane&15, M = v + 8*(lane>>4) (v = vgpr index 0..7)
    int mrow  = lane & 15;
    int khalf = (lane >> 4) << 1;         // 0 or 2
    int n0    = wave * 16;
    int ncol  = n0 + (lane & 15);

    const float* arow = atile + mrow * ATILE_STRIDE + khalf;
    const float* brow = Wt + ncol * DIM + khalf;

    v8f acc = {};
#pragma unroll
    for (int i = 0; i < 32; ++i) {        // K = 128 in steps of 4
        float2 ta = *(const float2*)(arow + 4 * i);   // ds_load_b64
        float2 tb = *(const float2*)(brow + 4 * i);   // global_load_b64 (L2-resident W)
        v2f av = {ta.x, ta.y};
        v2f bv = {tb.x, tb.y};
        acc = __builtin_amdgcn_wmma_f32_16x16x4_f32(
            /*neg_a=*/false, av, /*neg_b=*/false, bv,
            /*c_mod=*/(short)0, acc, /*reuse_a=*/false, /*reuse_b=*/false);
    }

    float bb = bias[ncol];
    int rbase = m0 + ((lane >> 4) << 3);
#pragma unroll
    for (int v = 0; v < 8; ++v) {
        float r = fmaxf(acc[v] + bb, 0.0f);
        io[(rbase + v) * DIM + ncol] = r;
    }
}

// ---------------------------------------------------------------------------
// Launch
// ---------------------------------------------------------------------------
extern "C" void kernel_launch(void* const* d_in, const int* in_sizes, int n_in,
                              void* d_out, int out_size, void* d_ws, size_t ws_size,
                              hipStream_t stream) {
    const float* x    = (const float*)d_in[0];   // [50000,128]
    const int*   ei   = (const int*)  d_in[1];   // [2,800000]
    const float* W    = (const float*)d_in[2];   // [128,128]
    const float* bias = (const float*)d_in[3];   // [128]
    float* out = (float*)d_out;                  // [50000,128]

    // workspace layout: deg_out/norm_src | deg_in/norm_dst | Wt  (~466 KB)
    float* deg_out = (float*)d_ws;
    float* deg_in  = deg_out + N_NODES;
    float* Wt      = deg_in + N_NODES;

    gcn_zero_kernel<<<(N_NODES * DIM + 255) / 256, 256, 0, stream>>>(out, deg_out);
    gcn_degree_kernel<<<(N_EDGES + 255) / 256, 256, 0, stream>>>(ei, deg_out, deg_in);
    gcn_norm_transpose_kernel<<<(N_NODES + 255) / 256, 256, 0, stream>>>(deg_out, deg_in, W, Wt);
    gcn_scatter_kernel<<<N_EDGES / 8, 256, 0, stream>>>(x, ei, deg_out, out);
    gcn_wmma_gemm_kernel<<<N_NODES / 16, 256, 0, stream>>>(Wt, bias, deg_in, out);
}